// nnmodel_35708358099045
// MI455X (gfx1250) — compile-verified
//
#include <hip/hip_runtime.h>
#include <hip/hip_bf16.h>

typedef float v2f __attribute__((ext_vector_type(2)));
typedef float v8f __attribute__((ext_vector_type(8)));

// ---------------- fused-weight element generators (graph matrices are fixed) ---------------
// A_ENC[j,i] = 1 if i in [4j-3, 4j+2] (mod 40); A_DEC[i,j] == A_ENC[j,i]
__device__ __forceinline__ float aenc(int j, int i) {
  int d = ((i - 4 * j + 3) % 40 + 40) % 40;
  return (d < 6) ? 1.0f : 0.0f;
}
// A_PRED[d,s] = (d==s) + w*(d==s-1 mod 10) + w*(d==s+1 mod 10), w = exp(-1/9)
__device__ __forceinline__ float apred(int d, int s) {
  const float wgt = 0.89483932f; // expf(-1.0f/9.0f)
  float v = 0.0f;
  if (d == s) v += 1.0f;
  if (d == (s + 9) % 10) v += wgt;
  if (d == (s + 1) % 10) v += wgt;
  return v;
}
// W1[(j,f), k]: k<40 -> A_ENC[j,k]*enc_rel_w[f]; k>=40 -> (j==j2)*enc_root_w[f,g]
__device__ __forceinline__ float w1_elem(int col, int k, const float* erw, const float* erootw) {
  int j = col >> 3, f = col & 7;
  if (k < 40) return aenc(j, k) * erw[f];
  int kc = k - 40, j2 = kc >> 3, g = kc & 7;
  return (j == j2) ? erootw[f * 8 + g] : 0.0f;
}
// W2[(j,f),(j2,g)] = A_PRED[j,j2]*pred_rel_w[f,g] + (j==j2)*pred_root_w[f,g]
__device__ __forceinline__ float w2_elem(int col, int k, const float* prw, const float* prootw) {
  int j = col >> 3, f = col & 7;
  int j2 = k >> 3, g = k & 7;
  float v = apred(j, j2) * prw[f * 8 + g];
  if (j == j2) v += prootw[f * 8 + g];
  return v;
}
// W3[i,(j,f)] = A_DEC[i,j]*dec_rel_w[f]; zero-padded rows i in [40,48)
__device__ __forceinline__ float w3_elem(int row, int k, const float* drw) {
  if (row >= 40) return 0.0f;
  int j = k >> 3, f = k & 7;
  return aenc(j, row) * drw[f];
}

// ---------------- workspace layout (floats) ----------------
// [0      , 9600 )  Bfrag1: 30 ksteps x 5 ntiles x 32 lanes x float2  (W1^T fragments)
// [9600   , 16000)  Bfrag2: 20 x 5 x 32 x float2                     (W2^T fragments)
// [16000  , 19840)  Bfrag3: 20 x 3 x 32 x float2                     (W3^T fragments, N padded to 48)
// [19840  , 19920)  b1[80]
// [19920  , 20000)  b2[80]
// [20000  , 20002)  {dec_rel_b, dec_root_w}
#define BF2OFF 9600
#define BF3OFF 16000
#define B1OFF  19840
#define B2OFF  19920
#define SCOFF  20000

__global__ void gnn_setup(const float* erw, const float* erb, const float* erootw,
                          const float* prw, const float* prb, const float* prootw,
                          const float* drw, const float* drb, const float* drootw,
                          float* ws) {
  const int NF1 = 30 * 5 * 32;   // 4800 float2 slots
  const int NF2 = 20 * 5 * 32;   // 3200
  const int NF3 = 20 * 3 * 32;   // 1920
  const int TB  = NF1 + NF2 + NF3;
  const int total = TB + 80 + 80 + 2;
  for (int t = blockIdx.x * blockDim.x + threadIdx.x; t < total;
       t += gridDim.x * blockDim.x) {
    if (t < NF1) {
      int lane = t & 31, rest = t >> 5;
      int n = rest % 5, kk = rest / 5;
      int col = n * 16 + (lane & 15);
      int k = 4 * kk + 2 * (lane >> 4);
      ws[2 * t]     = w1_elem(col, k,     erw, erootw);
      ws[2 * t + 1] = w1_elem(col, k + 1, erw, erootw);
    } else if (t < NF1 + NF2) {
      int s = t - NF1;
      int lane = s & 31, rest = s >> 5;
      int n = rest % 5, kk = rest / 5;
      int col = n * 16 + (lane & 15);
      int k = 4 * kk + 2 * (lane >> 4);
      ws[BF2OFF + 2 * s]     = w2_elem(col, k,     prw, prootw);
      ws[BF2OFF + 2 * s + 1] = w2_elem(col, k + 1, prw, prootw);
    } else if (t < TB) {
      int s = t - NF1 - NF2;
      int lane = s & 31, rest = s >> 5;
      int n = rest % 3, kk = rest / 3;
      int col = n * 16 + (lane & 15);
      int k = 4 * kk + 2 * (lane >> 4);
      ws[BF3OFF + 2 * s]     = w3_elem(col, k,     drw);
      ws[BF3OFF + 2 * s + 1] = w3_elem(col, k + 1, drw);
    } else {
      int c = t - TB;
      if (c < 80)        ws[B1OFF + c]        = erb[c & 7];
      else if (c < 160)  ws[B2OFF + (c - 80)] = prb[c & 7];
      else if (c == 160) ws[SCOFF]            = drb[0];
      else               ws[SCOFF + 1]        = drootw[0];
    }
  }
}

// ---------------- main WMMA kernel: one wave = 16 batch rows ----------------
__global__ __launch_bounds__(256) void gnn_wmma(
    const float* __restrict__ x, const float* __restrict__ z,
    const float* __restrict__ y, const float* __restrict__ ws,
    float* __restrict__ out, long long nTiles) {
  __shared__ float hbuf[8 * 16 * 84];  // per-wave 16x80 activations, stride 84 (bank-conflict free)
  const int lane  = threadIdx.x & 31;
  const int wid   = threadIdx.x >> 5;
  const long long gw = (long long)blockIdx.x * 8 + wid;
  if (gw >= nTiles) return;            // wave-uniform: EXEC stays all-1s for WMMA
  const int lcol  = lane & 15;         // N col within tile (D/C), M row (A)
  const int khalf = lane >> 4;         // selects K pair 0-1 vs 2-3 (A/B), M +8 (D/C)
  const long long row0 = gw * 16;
  const long long brow = row0 + lcol;

  const float2* bf1 = (const float2*)(ws);
  const float2* bf2 = (const float2*)(ws + BF2OFF);
  const float2* bf3 = (const float2*)(ws + BF3OFF);
  const float* b1v = ws + B1OFF;
  const float* b2v = ws + B2OFF;

  float* H = hbuf + wid * (16 * 84);

  v8f acc[5];

  // ======== Layer 1: [16x120] x [120x80], K = 30 steps of 4 ========
#pragma unroll
  for (int n = 0; n < 5; ++n) {
    float b = b1v[n * 16 + lcol];
    acc[n] = (v8f){b, b, b, b, b, b, b, b};
  }
#pragma unroll
  for (int kk = 0; kk < 30; ++kk) {
    const int k = 4 * kk + 2 * khalf;
    float2 av;
    if (4 * kk < 40) av = *(const float2*)(x + brow * 40 + k);
    else             av = *(const float2*)(z + brow * 80 + (k - 40));
    v2f a = {av.x, av.y};
#pragma unroll
    for (int n = 0; n < 5; ++n) {
      float2 bw = bf1[(kk * 5 + n) * 32 + lane];
      v2f b = {bw.x, bw.y};
      acc[n] = __builtin_amdgcn_wmma_f32_16x16x4_f32(
          false, a, false, b, (short)0, acc[n], false, false);
    }
  }
  // ReLU, D-layout -> row-major LDS tile
#pragma unroll
  for (int n = 0; n < 5; ++n)
#pragma unroll
    for (int r = 0; r < 8; ++r) {
      float v = acc[n][r];
      H[(r + 8 * khalf) * 84 + n * 16 + lcol] = v > 0.0f ? v : 0.0f;
    }

  // ======== Layer 2: [16x80] x [80x80], K = 20 steps of 4 ========
#pragma unroll
  for (int n = 0; n < 5; ++n) {
    float b = b2v[n * 16 + lcol];
    acc[n] = (v8f){b, b, b, b, b, b, b, b};
  }
#pragma unroll
  for (int kk = 0; kk < 20; ++kk) {
    const int k = 4 * kk + 2 * khalf;
    float2 hv = *(const float2*)(H + lcol * 84 + k);
    v2f a = {hv.x, hv.y};
#pragma unroll
    for (int n = 0; n < 5; ++n) {
      float2 bw = bf2[(kk * 5 + n) * 32 + lane];
      v2f b = {bw.x, bw.y};
      acc[n] = __builtin_amdgcn_wmma_f32_16x16x4_f32(
          false, a, false, b, (short)0, acc[n], false, false);
    }
  }
#pragma unroll
  for (int n = 0; n < 5; ++n)
#pragma unroll
    for (int r = 0; r < 8; ++r) {
      float v = acc[n][r];
      H[(r + 8 * khalf) * 84 + n * 16 + lcol] = v > 0.0f ? v : 0.0f;
    }

  // ======== Layer 3: [16x80] x [80x48(pad of 40)] ========
  v8f acc3[3];
#pragma unroll
  for (int n = 0; n < 3; ++n) acc3[n] = (v8f){0, 0, 0, 0, 0, 0, 0, 0};
#pragma unroll
  for (int kk = 0; kk < 20; ++kk) {
    const int k = 4 * kk + 2 * khalf;
    float2 hv = *(const float2*)(H + lcol * 84 + k);
    v2f a = {hv.x, hv.y};
#pragma unroll
    for (int n = 0; n < 3; ++n) {
      float2 bw = bf3[(kk * 3 + n) * 32 + lane];
      v2f b = {bw.x, bw.y};
      acc3[n] = __builtin_amdgcn_wmma_f32_16x16x4_f32(
          false, a, false, b, (short)0, acc3[n], false, false);
    }
  }
  const float bias3 = ws[SCOFF];
  const float cmul  = ws[SCOFF + 1];
#pragma unroll
  for (int n = 0; n < 3; ++n) {
    const int col = n * 16 + lcol;
    if (col < 40) {  // predicate only after all WMMA done
#pragma unroll
      for (int r = 0; r < 8; ++r) {
        long long orow = row0 + r + 8 * khalf;
        float yv = y[orow * 40 + col];
        out[orow * 40 + col] = acc3[n][r] + bias3 + cmul * yv;
      }
    }
  }
}

// ---------------- scalar tail for B % 16 rows ----------------
__global__ void gnn_tail(const float* x, const float* z, const float* y,
                         const float* erw, const float* erb, const float* erootw,
                         const float* prw, const float* prb, const float* prootw,
                         const float* drw, const float* drb, const float* drootw,
                         float* out, long long rowStart, int nrows) {
  int t = threadIdx.x;
  if (t >= nrows) return;
  long long b = rowStart + t;
  float in[120];
  for (int i = 0; i < 40; ++i) in[i] = x[b * 40 + i];
  for (int i = 0; i < 80; ++i) in[40 + i] = z[b * 80 + i];
  float h1[80];
  for (int c = 0; c < 80; ++c) {
    float s = erb[c & 7];
    for (int k = 0; k < 120; ++k) s += w1_elem(c, k, erw, erootw) * in[k];
    h1[c] = s > 0.0f ? s : 0.0f;
  }
  float h2[80];
  for (int c = 0; c < 80; ++c) {
    float s = prb[c & 7];
    for (int k = 0; k < 80; ++k) s += w2_elem(c, k, prw, prootw) * h1[k];
    h2[c] = s > 0.0f ? s : 0.0f;
  }
  for (int i = 0; i < 40; ++i) {
    float s = drb[0] + drootw[0] * y[b * 40 + i];
    for (int k = 0; k < 80; ++k) s += w3_elem(i, k, drw) * h2[k];
    out[b * 40 + i] = s;
  }
}

extern "C" void kernel_launch(void* const* d_in, const int* in_sizes, int n_in,
                              void* d_out, int out_size, void* d_ws, size_t ws_size,
                              hipStream_t stream) {
  const float* x        = (const float*)d_in[0];
  const float* z        = (const float*)d_in[1];
  const float* y        = (const float*)d_in[2];
  const float* erw      = (const float*)d_in[3];
  const float* erb      = (const float*)d_in[4];
  const float* erootw   = (const float*)d_in[5];
  const float* prw      = (const float*)d_in[6];
  const float* prb      = (const float*)d_in[7];
  const float* prootw   = (const float*)d_in[8];
  const float* drw      = (const float*)d_in[9];
  const float* drb      = (const float*)d_in[10];
  const float* drootw   = (const float*)d_in[11];
  float* out = (float*)d_out;
  float* ws  = (float*)d_ws;   // needs ~80 KB

  long long B = (long long)in_sizes[0] / 40;

  gnn_setup<<<40, 256, 0, stream>>>(erw, erb, erootw, prw, prb, prootw,
                                    drw, drb, drootw, ws);

  long long nTiles = B / 16;                 // 32768 waves for B=524288
  if (nTiles > 0) {
    long long blocks = (nTiles + 7) / 8;     // 8 waves (256 threads) per block
    gnn_wmma<<<dim3((unsigned)blocks), dim3(256), 0, stream>>>(x, z, y, ws, out, nTiles);
  }
  long long tail = B - nTiles * 16;
  if (tail > 0) {
    gnn_tail<<<1, 32, 0, stream>>>(x, z, y, erw, erb, erootw, prw, prb, prootw,
                                   drw, drb, drootw, out, nTiles * 16, (int)tail);
  }
}